// ProtoGraph_20005957665514
// MI455X (gfx1250) — compile-verified
//
#include <hip/hip_runtime.h>

typedef __attribute__((ext_vector_type(2))) float v2f;
typedef __attribute__((ext_vector_type(8))) float v8f;
typedef __attribute__((ext_vector_type(4))) int   v4i;

#define GAT_N 50000
#define GAT_E 800000
#define GAT_F 256
#define GAT_ETOT (GAT_E + GAT_N)
#define TILE_ROWS 32

#if defined(__gfx1250__) && __has_builtin(__builtin_amdgcn_global_load_async_to_lds_b128) && __has_builtin(__builtin_amdgcn_s_wait_asynccnt)
#define GAT_HAS_ASYNC 1
typedef __attribute__((address_space(1))) v4i gas_v4i;   // global
typedef __attribute__((address_space(3))) v4i las_v4i;   // LDS
#else
#define GAT_HAS_ASYNC 0
#endif

// ---------------------------------------------------------------------------
// GEMM: Y[nRows,256] = X[nRows,256] @ W[256,256] using V_WMMA_F32_16X16X4_F32
// Block: 256 threads = 8 waves. Each block: 32 rows x 256 cols.
// Wave w: row-tile rt = w>>2 (16 rows), col-quarter cq = w&3 (64 cols = 4 tiles).
// ---------------------------------------------------------------------------
__global__ __launch_bounds__(256) void k_gemm_wmma(const float* __restrict__ X,
                                                   const float* __restrict__ W,
                                                   float* __restrict__ Y,
                                                   int nRows) {
    __shared__ float xs[TILE_ROWS][260];   // pad to 260 floats: stride%64==4 -> conflict-free
    const int tid = threadIdx.x;
    const int rowBase = blockIdx.x * TILE_ROWS;
    const bool fullTile = (rowBase + TILE_ROWS) <= nRows;

#if GAT_HAS_ASYNC
    if (fullTile) {
        // Async DMA staging: memory -> LDS without touching VGPRs (ASYNCcnt).
        for (int i = tid; i < TILE_ROWS * 64; i += 256) {
            const int r  = i >> 6;
            const int c4 = i & 63;
            const float* g = X + (size_t)(rowBase + r) * GAT_F + c4 * 4;
            __builtin_amdgcn_global_load_async_to_lds_b128(
                (gas_v4i*)g, (las_v4i*)&xs[r][c4 * 4], 0, 0);
        }
        __builtin_amdgcn_s_wait_asynccnt(0);
    } else
#endif
    {
        for (int i = tid; i < TILE_ROWS * 64; i += 256) {
            const int r  = i >> 6;
            const int c4 = i & 63;
            float4 v = make_float4(0.f, 0.f, 0.f, 0.f);
            const int gr = rowBase + r;
            if (gr < nRows) v = *(const float4*)(X + (size_t)gr * GAT_F + c4 * 4);
            xs[r][c4 * 4 + 0] = v.x; xs[r][c4 * 4 + 1] = v.y;
            xs[r][c4 * 4 + 2] = v.z; xs[r][c4 * 4 + 3] = v.w;
        }
    }
    __syncthreads();

    const int lane  = tid & 31;
    const int wave  = tid >> 5;
    const int rt    = wave >> 2;            // 0..1
    const int cq    = wave & 3;             // 0..3
    const int mrow  = rt * 16 + (lane & 15);
    const int khalf = (lane >> 4) * 2;      // A frag: lanes 0-15 hold K=k,k+1; 16-31 hold K=k+2,k+3

    v8f acc[4] = {};

    for (int k = 0; k < 256; k += 4) {
        const int kk = k + khalf;
        v2f a;
        a.x = xs[mrow][kk];
        a.y = xs[mrow][kk + 1];
#pragma unroll
        for (int t = 0; t < 4; ++t) {
            const int n = cq * 64 + t * 16 + (lane & 15);
            v2f b;
            b.x = W[(size_t)kk * GAT_F + n];
            b.y = W[(size_t)(kk + 1) * GAT_F + n];
            acc[t] = __builtin_amdgcn_wmma_f32_16x16x4_f32(
                false, a, false, b, (short)0, acc[t], false, false);
        }
    }

    // C/D layout: VGPR r -> M = r + 8*(lane>=16), N = lane&15.
    const int rhi  = (lane >> 4) << 3;
    const int col0 = cq * 64 + (lane & 15);
    const int rb   = rowBase + rt * 16 + rhi;
    if (fullTile) {
#pragma unroll
        for (int t = 0; t < 4; ++t)
#pragma unroll
            for (int r = 0; r < 8; ++r)
                Y[(size_t)(rb + r) * GAT_F + col0 + t * 16] = acc[t][r];
    } else {
#pragma unroll
        for (int t = 0; t < 4; ++t)
#pragma unroll
            for (int r = 0; r < 8; ++r)
                if (rb + r < nRows)
                    Y[(size_t)(rb + r) * GAT_F + col0 + t * 16] = acc[t][r];
    }
}

// ---------------------------------------------------------------------------
// Per-node attention logits, one wave per node.  Lane l owns channels
// [8l, 8l+8); heads reduced with shfl_xor trees over g = C/8 lanes.
// ---------------------------------------------------------------------------
__global__ __launch_bounds__(256) void k_alpha(const float* __restrict__ xW,
                                               const float* __restrict__ a_src,
                                               const float* __restrict__ a_dst,
                                               float* __restrict__ as_out,
                                               float* __restrict__ ad_out,
                                               int n, int Hh, int Cc) {
    const int lane = threadIdx.x & 31;
    const int node = blockIdx.x * 8 + (threadIdx.x >> 5);
    if (node >= n) return;
    const int ch = lane * 8;
    const float* row = xW + (size_t)node * GAT_F + ch;
    const float4 x0 = *(const float4*)(row);
    const float4 x1 = *(const float4*)(row + 4);
    const float4 s0 = *(const float4*)(a_src + ch);
    const float4 s1 = *(const float4*)(a_src + ch + 4);
    const float4 d0 = *(const float4*)(a_dst + ch);
    const float4 d1 = *(const float4*)(a_dst + ch + 4);
    float s = x0.x*s0.x + x0.y*s0.y + x0.z*s0.z + x0.w*s0.w
            + x1.x*s1.x + x1.y*s1.y + x1.z*s1.z + x1.w*s1.w;
    float d = x0.x*d0.x + x0.y*d0.y + x0.z*d0.z + x0.w*d0.w
            + x1.x*d1.x + x1.y*d1.y + x1.z*d1.z + x1.w*d1.w;
    const int g = Cc >> 3;                 // lanes per head: 8 (H=4) or 32 (H=1)
#pragma unroll
    for (int m = 1; m < 32; m <<= 1) {
        if (m >= g) break;
        s += __shfl_xor(s, m, 32);
        d += __shfl_xor(d, m, 32);
    }
    if ((lane & (g - 1)) == 0) {
        const int h = lane / g;
        as_out[(size_t)node * Hh + h] = s;
        ad_out[(size_t)node * Hh + h] = d;
    }
}

// ---------------------------------------------------------------------------
// Init: acc = 0, m = -3e38, z = 0
// ---------------------------------------------------------------------------
__global__ __launch_bounds__(256) void k_init(float* __restrict__ acc,
                                              float* __restrict__ m,
                                              float* __restrict__ z,
                                              int accN, int mzN) {
    const int t = blockIdx.x * blockDim.x + threadIdx.x;
    if (t < accN) acc[t] = 0.f;
    if (t < mzN)  { m[t] = -3.0e38f; z[t] = 0.f; }
}

__device__ __forceinline__ void atomicMaxF(float* addr, float v) {
    // Sign-split monotonic mapping: correct for mixed-sign floats.
    if (v >= 0.f) atomicMax((int*)addr, __float_as_int(v));
    else          atomicMin((unsigned int*)addr, __float_as_uint(v));
}

__device__ __forceinline__ void edge_sd(const int* __restrict__ ei, long e, int& s, int& d) {
    if (e < GAT_E) { s = ei[e]; d = ei[GAT_E + e]; }
    else           { s = (int)(e - GAT_E); d = s; }   // self loop
}

// ---------------------------------------------------------------------------
// Edge pass 1: e = leaky_relu(as[src]+ad[dst]); store; segment max into m[dst]
// ---------------------------------------------------------------------------
__global__ __launch_bounds__(256) void k_edge_max(const int* __restrict__ ei,
                                                  const float* __restrict__ as_in,
                                                  const float* __restrict__ ad_in,
                                                  float* __restrict__ m,
                                                  float* __restrict__ wbuf,
                                                  int Hh) {
    const long e = (long)blockIdx.x * blockDim.x + threadIdx.x;
    if (e >= GAT_ETOT) return;
    int s, d; edge_sd(ei, e, s, d);
    for (int h = 0; h < Hh; ++h) {
        float v = as_in[(size_t)s * Hh + h] + ad_in[(size_t)d * Hh + h];
        v = v > 0.f ? v : 0.2f * v;
        wbuf[(size_t)e * Hh + h] = v;
        atomicMaxF(&m[(size_t)d * Hh + h], v);
    }
}

// ---------------------------------------------------------------------------
// Edge pass 2: w = exp(e - m[dst]); store; segment sum into z[dst]
// ---------------------------------------------------------------------------
__global__ __launch_bounds__(256) void k_edge_expsum(const int* __restrict__ ei,
                                                     float* __restrict__ wbuf,
                                                     const float* __restrict__ m,
                                                     float* __restrict__ z,
                                                     int Hh) {
    const long e = (long)blockIdx.x * blockDim.x + threadIdx.x;
    if (e >= GAT_ETOT) return;
    int s, d; edge_sd(ei, e, s, d);
    (void)s;
    for (int h = 0; h < Hh; ++h) {
        const float w = __expf(wbuf[(size_t)e * Hh + h] - m[(size_t)d * Hh + h]);
        wbuf[(size_t)e * Hh + h] = w;
        atomicAdd(&z[(size_t)d * Hh + h], w);
    }
}

// ---------------------------------------------------------------------------
// Edge pass 3: acc[dst] += xW[src] * (w / z[dst]).  One wave per edge,
// each lane owns 8 contiguous channels (float4 x2 gather, 8 atomic f32 adds).
// ---------------------------------------------------------------------------
__global__ __launch_bounds__(256) void k_edge_aggr(const int* __restrict__ ei,
                                                   const float* __restrict__ xW,
                                                   const float* __restrict__ wbuf,
                                                   const float* __restrict__ z,
                                                   float* __restrict__ acc,
                                                   int Hh, int Cc) {
    const int lane = threadIdx.x & 31;
    const int wv   = threadIdx.x >> 5;
    const long e   = (long)blockIdx.x * 8 + wv;
    if (e >= GAT_ETOT) return;
    int s, d; edge_sd(ei, e, s, d);

    const int ch = lane * 8;          // 256 channels / 32 lanes
    const int h  = ch / Cc;
    const float alpha = wbuf[(size_t)e * Hh + h] / z[(size_t)d * Hh + h];

    const float* xr = xW  + (size_t)s * GAT_F + ch;
    float*       ar = acc + (size_t)d * GAT_F + ch;
    __builtin_prefetch(ar, 1, 1);     // global_prefetch on gfx1250

    const float4 v0 = *(const float4*)(xr);
    const float4 v1 = *(const float4*)(xr + 4);
    atomicAdd(ar + 0, v0.x * alpha);
    atomicAdd(ar + 1, v0.y * alpha);
    atomicAdd(ar + 2, v0.z * alpha);
    atomicAdd(ar + 3, v0.w * alpha);
    atomicAdd(ar + 4, v1.x * alpha);
    atomicAdd(ar + 5, v1.y * alpha);
    atomicAdd(ar + 6, v1.z * alpha);
    atomicAdd(ar + 7, v1.w * alpha);
}

// ---------------------------------------------------------------------------
// Bias (+ optional ELU), in place.  256-column row-major -> col = t & 255.
// ---------------------------------------------------------------------------
__global__ __launch_bounds__(256) void k_bias_act(float* __restrict__ acc,
                                                  const float* __restrict__ bias,
                                                  int total, int applyElu) {
    const int t = blockIdx.x * blockDim.x + threadIdx.x;
    if (t >= total) return;
    float v = acc[t] + bias[t & (GAT_F - 1)];
    if (applyElu) v = v > 0.f ? v : (__expf(v) - 1.f);
    acc[t] = v;
}

// ---------------------------------------------------------------------------
extern "C" void kernel_launch(void* const* d_in, const int* in_sizes, int n_in,
                              void* d_out, int out_size, void* d_ws, size_t ws_size,
                              hipStream_t stream) {
    (void)in_sizes; (void)n_in; (void)out_size; (void)ws_size;

    const float* x      = (const float*)d_in[0];
    const int*   ei     = (const int*)  d_in[1];
    const float* W1     = (const float*)d_in[2];
    const float* a_src1 = (const float*)d_in[3];
    const float* a_dst1 = (const float*)d_in[4];
    const float* b1     = (const float*)d_in[5];
    const float* W2     = (const float*)d_in[6];
    const float* a_src2 = (const float*)d_in[7];
    const float* a_dst2 = (const float*)d_in[8];
    const float* b2     = (const float*)d_in[9];
    float* out = (float*)d_out;

    const int N = GAT_N, H = 4, C = 64;
    const size_t NF = (size_t)N * GAT_F;

    // Workspace carve-up (floats).  Peak ~122 MB.
    float* ws   = (float*)d_ws;
    size_t o    = 0;
    float* bufA = ws + o; o += NF;               // xW (both layers)
    float* bufB = ws + o; o += NF;               // acc1 -> h (layer-2 input)
    float* as_  = ws + o; o += (size_t)N * H;
    float* ad_  = ws + o; o += (size_t)N * H;
    float* m_   = ws + o; o += (size_t)N * H;
    float* z_   = ws + o; o += (size_t)N * H;
    float* wbuf = ws + o; o += (size_t)GAT_ETOT * H;

    const int gemmBlocks  = (N + TILE_ROWS - 1) / TILE_ROWS;
    const int alphaBlocks = (N + 7) / 8;          // one wave per node
    const int edgeBlocks  = (GAT_ETOT + 255) / 256;
    const int aggrBlocks  = (GAT_ETOT + 7) / 8;
    const int nfBlocks    = (int)((NF + 255) / 256);

    // ===================== Layer 1 (H=4, C=64) =====================
    k_gemm_wmma  <<<gemmBlocks, 256, 0, stream>>>(x, W1, bufA, N);
    k_alpha      <<<alphaBlocks, 256, 0, stream>>>(bufA, a_src1, a_dst1, as_, ad_, N, H, C);
    k_init       <<<nfBlocks, 256, 0, stream>>>(bufB, m_, z_, (int)NF, N * H);
    k_edge_max   <<<edgeBlocks, 256, 0, stream>>>(ei, as_, ad_, m_, wbuf, H);
    k_edge_expsum<<<edgeBlocks, 256, 0, stream>>>(ei, wbuf, m_, z_, H);
    k_edge_aggr  <<<aggrBlocks, 256, 0, stream>>>(ei, bufA, wbuf, z_, bufB, H, C);
    k_bias_act   <<<nfBlocks, 256, 0, stream>>>(bufB, b1, (int)NF, 1);   // ELU

    // ===================== Layer 2 (H=1, C=256) ====================
    k_gemm_wmma  <<<gemmBlocks, 256, 0, stream>>>(bufB, W2, bufA, N);
    k_alpha      <<<alphaBlocks, 256, 0, stream>>>(bufA, a_src2, a_dst2, as_, ad_, N, 1, GAT_F);
    k_init       <<<nfBlocks, 256, 0, stream>>>(out, m_, z_, (int)NF, N);
    k_edge_max   <<<edgeBlocks, 256, 0, stream>>>(ei, as_, ad_, m_, wbuf, 1);
    k_edge_expsum<<<edgeBlocks, 256, 0, stream>>>(ei, wbuf, m_, z_, 1);
    k_edge_aggr  <<<aggrBlocks, 256, 0, stream>>>(ei, bufA, wbuf, z_, out, 1, GAT_F);
    k_bias_act   <<<nfBlocks, 256, 0, stream>>>(out, b2, (int)NF, 0);    // bias only
}